// LabelGraphClassifier_21182778704610
// MI455X (gfx1250) — compile-verified
//
#include <hip/hip_runtime.h>

typedef __attribute__((ext_vector_type(2))) float v2f;
typedef __attribute__((ext_vector_type(8))) float v8f;

#define IN_F 128
#define OUT_F 32
#define WPB 4      // waves per block in the GEMM
#define LDA 132    // padded LDS row stride for A tile (132 mod 64 banks = 4 -> conflict free)
#define LDW 132    // padded LDS row stride for transposed W

// ---------------------------------------------------------------------------
// Degree init: self-loop contributes 1 to both in- and out-degree.
// ---------------------------------------------------------------------------
__global__ void gcn_deg_init(float* __restrict__ dout, float* __restrict__ din, int n) {
  const int i = blockIdx.x * blockDim.x + threadIdx.x;
  if (i < n) { dout[i] = 1.0f; din[i] = 1.0f; }
}

__global__ void gcn_deg_count(const int* __restrict__ src, const int* __restrict__ dst,
                              float* __restrict__ dout, float* __restrict__ din, int e) {
  const int i = blockIdx.x * blockDim.x + threadIdx.x;
  if (i < e) {
    unsafeAtomicAdd(&dout[src[i]], 1.0f);
    unsafeAtomicAdd(&din[dst[i]], 1.0f);
  }
}

// ---------------------------------------------------------------------------
// h = (feature * rsqrt(out_deg)) @ W via V_WMMA_F32_16X16X4_F32.
// One wave owns a 16-row M tile and both 16-wide N tiles (OUT_F = 32).
// Also seeds rst (= d_out) with h: the self-loop message with weight 1.0.
// ---------------------------------------------------------------------------
__global__ __launch_bounds__(WPB * 32) void gcn_gemm(
    const float* __restrict__ feature, const float* __restrict__ W,
    const float* __restrict__ deg_out, float* __restrict__ h,
    float* __restrict__ rst, int n)
{
  __shared__ float Wt[OUT_F * LDW];        // W transposed: Wt[nOut][k]
  __shared__ float At[WPB][16 * LDA];      // per-wave A tile

  const int lane = threadIdx.x & 31;
  const int wave = threadIdx.x >> 5;

  // Stage W (128x32, k-major in memory) transposed into LDS, n-major / k-inner,
  // so each lane's B fragment is a contiguous float2.
  for (int idx = threadIdx.x; idx < IN_F * OUT_F; idx += WPB * 32) {
    const int k  = idx >> 5;   // idx / OUT_F
    const int nn = idx & 31;   // idx % OUT_F
    Wt[nn * LDW + k] = W[idx];
  }
  __syncthreads();

  const int ntiles = (n + 15) >> 4;
  const int tile   = blockIdx.x * WPB + wave;
  if (tile >= ntiles) return;              // wave-uniform: EXEC stays all-ones
  const int m0 = tile << 4;

  // Coalesced load of the 16x128 A tile: each lane streams float4 columns.
  float* A = &At[wave][0];
  {
    const int c = lane << 2;               // 32 lanes * 4 floats = 128 cols
    for (int r = 0; r < 16; ++r) {
      int row = m0 + r; if (row >= n) row = n - 1;
      const float4 v = *(const float4*)(feature + (size_t)row * IN_F + c);
      A[r * LDA + c + 0] = v.x;
      A[r * LDA + c + 1] = v.y;
      A[r * LDA + c + 2] = v.z;
      A[r * LDA + c + 3] = v.w;
    }
  }

  const int ln = lane & 15;                // M for A frag, N for B/C/D frags
  const int hi = lane >> 4;                // selects K pair {2hi, 2hi+1}
  int mrow = m0 + ln; if (mrow >= n) mrow = n - 1;
  const float rs = rsqrtf(deg_out[mrow]);  // fold the degree norm into A

  v8f c0 = {}; v8f c1 = {};
  const float* Arow = A  + ln * LDA + 2 * hi;
  const float* B0   = Wt + ln * LDW + 2 * hi;
  const float* B1   = Wt + (ln + 16) * LDW + 2 * hi;
#pragma unroll
  for (int k0 = 0; k0 < IN_F; k0 += 4) {
    v2f a, b0, b1;
    a.x  = Arow[k0] * rs;  a.y  = Arow[k0 + 1] * rs;
    b0.x = B0[k0];         b0.y = B0[k0 + 1];
    b1.x = B1[k0];         b1.y = B1[k0 + 1];
    c0 = __builtin_amdgcn_wmma_f32_16x16x4_f32(false, a, false, b0, (short)0, c0, false, false);
    c1 = __builtin_amdgcn_wmma_f32_16x16x4_f32(false, a, false, b1, (short)0, c1, false, false);
  }

  // C/D layout: VGPR r -> row (r + 8*hi), col = ln (per ISA 16x16 f32 table).
#pragma unroll
  for (int r = 0; r < 8; ++r) {
    const int row = m0 + r + 8 * hi;
    if (row < n) {
      const size_t o = (size_t)row * OUT_F + ln;
      const float v0 = c0[r], v1 = c1[r];
      h[o]        = v0;  h[o + 16]   = v1;
      rst[o]      = v0;  rst[o + 16] = v1;   // self-loop message (weight 1.0)
    }
  }
}

// ---------------------------------------------------------------------------
// Edge scatter: 32 lanes per edge (one feature each) -> the gather of
// h[src]*32 floats and the 32 atomics per edge are each one contiguous
// 128-byte burst. h and rst are ~12.8 MB each: resident in the 192 MB L2.
// ---------------------------------------------------------------------------
__global__ void gcn_scatter(const int* __restrict__ src, const int* __restrict__ dst,
                            const float* __restrict__ ew, const float* __restrict__ h,
                            float* __restrict__ rst, int e)
{
  const long long gid = (long long)blockIdx.x * blockDim.x + threadIdx.x;
  const int ed = (int)(gid >> 5);
  const int f  = (int)(gid & 31);
  if (ed < e) {
    const float v = h[(size_t)src[ed] * OUT_F + f] * ew[ed];
    unsafeAtomicAdd(&rst[(size_t)dst[ed] * OUT_F + f], v);
  }
}

// ---------------------------------------------------------------------------
// rst = rst * rsqrt(in_deg) + b, in place on d_out.
// ---------------------------------------------------------------------------
__global__ void gcn_finalize(const float* __restrict__ din, const float* __restrict__ b,
                             float* __restrict__ out, int n)
{
  const long long gid = (long long)blockIdx.x * blockDim.x + threadIdx.x;
  const int i = (int)(gid >> 5);
  const int f = (int)(gid & 31);
  if (i < n) {
    const size_t o = (size_t)i * OUT_F + f;
    out[o] = out[o] * rsqrtf(din[i]) + b[f];
  }
}

// ---------------------------------------------------------------------------
extern "C" void kernel_launch(void* const* d_in, const int* in_sizes, int n_in,
                              void* d_out, int out_size, void* d_ws, size_t ws_size,
                              hipStream_t stream) {
  const float* feature = (const float*)d_in[0];
  const float* edge_w  = (const float*)d_in[1];
  const float* W       = (const float*)d_in[2];
  const float* b       = (const float*)d_in[3];
  const int*   src     = (const int*)d_in[4];
  const int*   dst     = (const int*)d_in[5];

  const int N = in_sizes[0] / IN_F;
  const int E = in_sizes[1];

  // Workspace layout (floats): [deg_out N][deg_in N][h N*OUT_F]; rst lives in d_out.
  float* deg_out = (float*)d_ws;
  float* deg_in  = deg_out + N;
  float* h       = deg_in + N;
  float* rst     = (float*)d_out;

  gcn_deg_init<<<(N + 255) / 256, 256, 0, stream>>>(deg_out, deg_in, N);
  gcn_deg_count<<<(E + 255) / 256, 256, 0, stream>>>(src, dst, deg_out, deg_in, E);

  const int ntiles = (N + 15) / 16;
  gcn_gemm<<<(ntiles + WPB - 1) / WPB, WPB * 32, 0, stream>>>(feature, W, deg_out, h, rst, N);

  const long long sthreads = (long long)E * 32;
  gcn_scatter<<<(unsigned)((sthreads + 255) / 256), 256, 0, stream>>>(src, dst, edge_w, h, rst, E);

  const long long fthreads = (long long)N * 32;
  gcn_finalize<<<(unsigned)((fthreads + 255) / 256), 256, 0, stream>>>(deg_in, b, (float*)d_out, N);
}